// PNA_Custom_72232759984527
// MI455X (gfx1250) — compile-verified
//
#include <hip/hip_runtime.h>

#define NN 100000
#define NE 800000
#define FD 64
#define NLAYER 3

typedef __attribute__((ext_vector_type(2))) float v2f;
typedef __attribute__((ext_vector_type(8))) float v8f;

// D = A(16x4,f32) * B(4x16,f32) + C(16x16,f32)  -- native fp32 WMMA on gfx1250
static __device__ __forceinline__ v8f wmma_f32(v2f a, v2f b, v8f c) {
  return __builtin_amdgcn_wmma_f32_16x16x4_f32(false, a, false, b, (short)0, c,
                                               false, false);
}

// ---------------------------------------------------------------- utilities
__global__ void k_zero(int* __restrict__ p, int n) {
  int i = blockIdx.x * 256 + threadIdx.x;
  if (i < n) p[i] = 0;
}

__global__ void k_count(const int* __restrict__ edst, int* __restrict__ cnt) {
  int e = blockIdx.x * 256 + threadIdx.x;
  if (e < NE) atomicAdd(&cnt[edst[e]], 1);
}

// single-block exclusive scan of cnt[NN] -> rowstart[NN+1]
__global__ __launch_bounds__(1024) void k_scan(const int* __restrict__ cnt,
                                               int* __restrict__ rowstart) {
  __shared__ int sh[1024];
  const int T = 1024;
  const int CH = (NN + T - 1) / T;
  const int t = threadIdx.x;
  const int beg = t * CH;
  const int end = (beg + CH < NN) ? (beg + CH) : NN;
  int s = 0;
  for (int i = beg; i < end; ++i) s += cnt[i];
  sh[t] = s;
  __syncthreads();
  for (int o = 1; o < T; o <<= 1) {
    int v = (t >= o) ? sh[t - o] : 0;
    __syncthreads();
    sh[t] += v;
    __syncthreads();
  }
  int run = sh[t] - s;  // exclusive prefix of this chunk
  for (int i = beg; i < end; ++i) { rowstart[i] = run; run += cnt[i]; }
  if (t == T - 1) rowstart[NN] = sh[T - 1];
}

__global__ void k_scatter(const int* __restrict__ edst,
                          const int* __restrict__ rowstart,
                          int* __restrict__ cursor, int* __restrict__ pos) {
  int e = blockIdx.x * 256 + threadIdx.x;
  if (e < NE) {
    int d = edst[e];
    int o = atomicAdd(&cursor[d], 1);
    pos[e] = rowstart[d] + o;
  }
}

__global__ void k_avglog(const float* __restrict__ hist, float* __restrict__ out) {
  if (blockIdx.x == 0 && threadIdx.x == 0) {
    float s = 0.f, w = 0.f;
    for (int i = 0; i < 33; ++i) { s += logf((float)i + 1.f) * hist[i]; w += hist[i]; }
    *out = s / w;
  }
}

// Re-pack row-major W[K][64] into paired-k layout so a B fragment
// (elements (k,n),(k+1,n)) is one aligned float2 load:
//   P[(k>>1)*128 + 2*n + (k&1)] = W[k*64+n]
__global__ void k_pair(const float* __restrict__ W, float* __restrict__ P, int K) {
  int i = blockIdx.x * 256 + threadIdx.x;
  if (i < K * 64) {
    int k = i >> 6, n = i & 63;
    P[((size_t)(k >> 1) * 128) + 2 * n + (k & 1)] = W[i];
  }
}

// Generic A-region GEMM: accumulates [16 x KSTEPS*4] @ paired-B into 4 N-tiles.
template <int KSTEPS, bool SCALE>
static __device__ __forceinline__ void gemm_region(
    const float* __restrict__ arow, float scale, const float* __restrict__ bp,
    int half, int lan, v8f acc[4]) {
#pragma unroll
  for (int kb = 0; kb < KSTEPS * 4; kb += 4) {
    const int ks = kb + 2 * half;
    v2f a = *(const v2f*)(arow + ks);
    if (SCALE) { a.x *= scale; a.y *= scale; }
#pragma unroll
    for (int t = 0; t < 4; ++t) {
      v2f b = *(const v2f*)(bp + (size_t)(ks >> 1) * 128 + 2 * (t * 16 + lan));
      acc[t] = wmma_f32(a, b, acc[t]);
    }
  }
}

// ------------------------------------------------- message GEMM (per layer)
// Per wave: 16 edges, A = cat[x[dst], x[src]] (16x128), B = W_pre (128x64, LDS),
// output rows scattered into dst-sorted hbuf[pos[e]].
__global__ __launch_bounds__(256) void k_msg(
    const float* __restrict__ x, const int* __restrict__ esrc,
    const int* __restrict__ edst, const int* __restrict__ pos,
    const float* __restrict__ wpre_p, const float* __restrict__ bpre,
    float* __restrict__ hbuf) {
  __shared__ float Bs[64 * 160];  // 64 k-pairs, pitch 160 dwords -> conflict-free
  for (int i = threadIdx.x; i < 128 * 64; i += 256) {
    int p = i >> 7, c = i & 127;
    Bs[p * 160 + c] = wpre_p[i];
  }
  __syncthreads();
  const int wave = threadIdx.x >> 5, lane = threadIdx.x & 31;
  const int half = lane >> 4, lan = lane & 15;
  const int tile = blockIdx.x * 8 + wave;  // grid = NE/128 exactly
  const int e = tile * 16 + lan;
  const int de = edst[e], se = esrc[e];
  const float* xd = x + (size_t)de * FD;
  const float* xs = x + (size_t)se * FD;
  v8f acc[4] = {};
#pragma unroll
  for (int kb = 0; kb < 128; kb += 4) {
    const int ks = kb + 2 * half;
    const float* ap = (kb < 64) ? (xd + ks) : (xs + (ks - 64));
    v2f a = *(const v2f*)ap;
#pragma unroll
    for (int t = 0; t < 4; ++t) {
      v2f b = *(const v2f*)(&Bs[(ks >> 1) * 160 + 2 * (t * 16 + lan)]);
      acc[t] = wmma_f32(a, b, acc[t]);
    }
  }
  int prow[8];
#pragma unroll
  for (int r = 0; r < 8; ++r) prow[r] = pos[tile * 16 + r + 8 * half];
#pragma unroll
  for (int t = 0; t < 4; ++t) {
    const float bias = bpre[t * 16 + lan];
#pragma unroll
    for (int r = 0; r < 8; ++r)
      hbuf[(size_t)prow[r] * FD + t * 16 + lan] = acc[t][r] + bias;
  }
}

// -------------------------------------- per-node aggregation (wave per node)
__global__ __launch_bounds__(256) void k_agg(
    const float* __restrict__ hbuf, const int* __restrict__ rowstart,
    const float* __restrict__ avglog_p, float* __restrict__ agg,
    float* __restrict__ s1, float* __restrict__ s2) {
  const int wave = threadIdx.x >> 5, lane = threadIdx.x & 31;
  const int n = blockIdx.x * 8 + wave;  // grid = NN/8 exactly
  const int beg = rowstart[n], end = rowstart[n + 1];
  const int cnt = end - beg;
  const float deg = (cnt > 0) ? (float)cnt : 1.f;
  v2f sum = {0.f, 0.f}, sq = {0.f, 0.f};
  v2f mn = {INFINITY, INFINITY}, mx = {-INFINITY, -INFINITY};
  for (int r = beg; r < end; ++r) {
    v2f v = *(const v2f*)(hbuf + (size_t)r * FD + 2 * lane);
    sum += v;
    sq += v * v;
    mn.x = fminf(mn.x, v.x); mn.y = fminf(mn.y, v.y);
    mx.x = fmaxf(mx.x, v.x); mx.y = fmaxf(mx.y, v.y);
  }
  if (cnt == 0) { mn = (v2f){0.f, 0.f}; mx = (v2f){0.f, 0.f}; }
  v2f mean = sum / deg;
  v2f msq = sq / deg;
  v2f sd;
  sd.x = sqrtf(fmaxf(msq.x - mean.x * mean.x, 0.f) + 1e-5f);
  sd.y = sqrtf(fmaxf(msq.y - mean.y * mean.y, 0.f) + 1e-5f);
  float* ap = agg + (size_t)n * 256;
  *(v2f*)(ap + 2 * lane) = mean;
  *(v2f*)(ap + 64 + 2 * lane) = mn;
  *(v2f*)(ap + 128 + 2 * lane) = mx;
  *(v2f*)(ap + 192 + 2 * lane) = sd;
  if (lane == 0) {
    const float al = *avglog_p;
    const float logd = logf(deg + 1.f);
    s1[n] = logd / al;
    s2[n] = al / logd;
  }
}

// ----------- fused post-NN GEMM (832->64) + lin GEMM (64->64) + res + ELU
__global__ __launch_bounds__(256) void k_post(
    const float* __restrict__ xin, const float* __restrict__ agg,
    const float* __restrict__ s1, const float* __restrict__ s2,
    const float* __restrict__ wpost_p, const float* __restrict__ bpost,
    const float* __restrict__ wlin_p, const float* __restrict__ blin,
    float* __restrict__ xout) {
  __shared__ float tls[8][16 * 68 + 8];  // per-wave 16x64 staging, pitch 68
  const int wave = threadIdx.x >> 5, lane = threadIdx.x & 31;
  const int half = lane >> 4, lan = lane & 15;
  const int tile = blockIdx.x * 8 + wave;
  if (tile >= NN / 16) return;  // no __syncthreads below
  const int node = tile * 16 + lan;
  const float s1v = s1[node], s2v = s2[node];
  const float* xr = xin + (size_t)node * FD;
  const float* ar = agg + (size_t)node * 256;
  v8f acc[4] = {};
  gemm_region<16, false>(xr, 1.f, wpost_p, half, lan, acc);              // x
  gemm_region<64, false>(ar, 1.f, wpost_p + 32 * 128, half, lan, acc);   // agg
  gemm_region<64, true>(ar, s1v, wpost_p + 160 * 128, half, lan, acc);   // agg*s1
  gemm_region<64, true>(ar, s2v, wpost_p + 288 * 128, half, lan, acc);   // agg*s2
  // stage intermediate (16x64) + bias into LDS for the second GEMM's A side
  float* tl = tls[wave];
#pragma unroll
  for (int t = 0; t < 4; ++t) {
    const float bias = bpost[t * 16 + lan];
#pragma unroll
    for (int r = 0; r < 8; ++r)
      tl[(r + 8 * half) * 68 + t * 16 + lan] = acc[t][r] + bias;
  }
  v8f acc2[4] = {};
#pragma unroll
  for (int kb = 0; kb < 64; kb += 4) {
    const int ks = kb + 2 * half;
    v2f a = *(const v2f*)(tl + lan * 68 + ks);
#pragma unroll
    for (int t = 0; t < 4; ++t) {
      v2f b = *(const v2f*)(wlin_p + (size_t)(ks >> 1) * 128 + 2 * (t * 16 + lan));
      acc2[t] = wmma_f32(a, b, acc2[t]);
    }
  }
#pragma unroll
  for (int t = 0; t < 4; ++t) {
    const float bias = blin[t * 16 + lan];
#pragma unroll
    for (int r = 0; r < 8; ++r) {
      const int m = tile * 16 + r + 8 * half;
      const int c = t * 16 + lan;
      float v = acc2[t][r] + bias + xin[(size_t)m * FD + c];
      v = (v > 0.f) ? v : expm1f(v);  // ELU(alpha=1)
      xout[(size_t)m * FD + c] = v;
    }
  }
}

// --------------------------------------- jumping-knowledge GEMM (192 -> 64)
__global__ __launch_bounds__(256) void k_jk(
    const float* __restrict__ x0, const float* __restrict__ x1,
    const float* __restrict__ x2, const float* __restrict__ wjk_p,
    const float* __restrict__ bjk, float* __restrict__ out) {
  const int wave = threadIdx.x >> 5, lane = threadIdx.x & 31;
  const int half = lane >> 4, lan = lane & 15;
  const int tile = blockIdx.x * 8 + wave;
  if (tile >= NN / 16) return;
  const int node = tile * 16 + lan;
  v8f acc[4] = {};
  gemm_region<16, false>(x0 + (size_t)node * FD, 1.f, wjk_p, half, lan, acc);
  gemm_region<16, false>(x1 + (size_t)node * FD, 1.f, wjk_p + 32 * 128, half, lan, acc);
  gemm_region<16, false>(x2 + (size_t)node * FD, 1.f, wjk_p + 64 * 128, half, lan, acc);
#pragma unroll
  for (int t = 0; t < 4; ++t) {
    const float bias = bjk[t * 16 + lan];
#pragma unroll
    for (int r = 0; r < 8; ++r) {
      const int m = tile * 16 + r + 8 * half;
      out[(size_t)m * FD + t * 16 + lan] = acc[t][r] + bias;
    }
  }
}

// ---------------------------------------------------------------- launcher
extern "C" void kernel_launch(void* const* d_in, const int* in_sizes, int n_in,
                              void* d_out, int out_size, void* d_ws,
                              size_t ws_size, hipStream_t stream) {
  (void)in_sizes; (void)n_in; (void)out_size; (void)ws_size;
  const float* x = (const float*)d_in[0];
  const int* ei = (const int*)d_in[1];
  const int* esrc = ei;
  const int* edst = ei + NE;
  const float* hist = (const float*)d_in[2];
  const float* Wpre = (const float*)d_in[3];
  const float* bpre = (const float*)d_in[4];
  const float* Wpost = (const float*)d_in[5];
  const float* bpost = (const float*)d_in[6];
  const float* Wlin = (const float*)d_in[7];
  const float* blin = (const float*)d_in[8];
  const float* Wjk = (const float*)d_in[9];
  const float* bjk = (const float*)d_in[10];
  float* out = (float*)d_out;

  char* w = (char*)d_ws;
  size_t off = 0;
  auto take = [&](size_t bytes) -> char* {
    char* p = w + off;
    off = (off + bytes + 255) & ~(size_t)255;
    return p;
  };
  int* cnt = (int*)take((size_t)NN * 4);
  int* rowstart = (int*)take((size_t)(NN + 1) * 4);
  int* cursor = (int*)take((size_t)NN * 4);
  int* pos = (int*)take((size_t)NE * 4);
  float* avglog = (float*)take(256);
  float* s1 = (float*)take((size_t)NN * 4);
  float* s2 = (float*)take((size_t)NN * 4);
  float* wpre_p = (float*)take((size_t)128 * 64 * 4);
  float* wpost_p = (float*)take((size_t)832 * 64 * 4);
  float* wlin_p = (float*)take((size_t)64 * 64 * 4);
  float* wjk_p = (float*)take((size_t)192 * 64 * 4);
  float* aggb = (float*)take((size_t)NN * 256 * 4);
  float* hbuf = (float*)take((size_t)NE * 64 * 4);
  float* xb[3];
  for (int i = 0; i < 3; ++i) xb[i] = (float*)take((size_t)NN * 64 * 4);

  // one-time CSR build (edge_index constant across layers)
  k_zero<<<(NN + 255) / 256, 256, 0, stream>>>(cnt, NN);
  k_zero<<<(NN + 255) / 256, 256, 0, stream>>>(cursor, NN);
  k_count<<<(NE + 255) / 256, 256, 0, stream>>>(edst, cnt);
  k_scan<<<1, 1024, 0, stream>>>(cnt, rowstart);
  k_scatter<<<(NE + 255) / 256, 256, 0, stream>>>(edst, rowstart, cursor, pos);
  k_avglog<<<1, 64, 0, stream>>>(hist, avglog);
  k_pair<<<(192 * 64 + 255) / 256, 256, 0, stream>>>(Wjk, wjk_p, 192);

  const float* xin = x;
  for (int l = 0; l < NLAYER; ++l) {
    k_pair<<<(128 * 64 + 255) / 256, 256, 0, stream>>>(Wpre + (size_t)l * 128 * 64, wpre_p, 128);
    k_pair<<<(832 * 64 + 255) / 256, 256, 0, stream>>>(Wpost + (size_t)l * 832 * 64, wpost_p, 832);
    k_pair<<<(64 * 64 + 255) / 256, 256, 0, stream>>>(Wlin + (size_t)l * 64 * 64, wlin_p, 64);
    k_msg<<<NE / 128, 256, 0, stream>>>(xin, esrc, edst, pos, wpre_p, bpre + l * 64, hbuf);
    k_agg<<<NN / 8, 256, 0, stream>>>(hbuf, rowstart, avglog, aggb, s1, s2);
    k_post<<<(NN / 16 + 7) / 8, 256, 0, stream>>>(xin, aggb, s1, s2, wpost_p,
                                                  bpost + l * 64, wlin_p,
                                                  blin + l * 64, xb[l]);
    xin = xb[l];
  }
  k_jk<<<(NN / 16 + 7) / 8, 256, 0, stream>>>(xb[0], xb[1], xb[2], wjk_p, bjk, out);
}